// Hamiltonian2PauliString_31138512896529
// MI455X (gfx1250) — compile-verified
//
#include <hip/hip_runtime.h>

// ---------------------------------------------------------------------------
// 2^24-point Walsh-Hadamard transform / 2^24 for MI455X (gfx1250).
//
// y[k] = (1/N) sum_j (-1)^popcount(k&j) x[j] factors into six H16 transforms
// (one per hex digit, order independent).  Each H16 application on a 16x16
// tile = 16x16x16 f32 matmul with a +/-1 matrix -> chained
// V_WMMA_F32_16X16X4_F32 (exact f32 WMMA, 4 K-chunks, H generated in-reg).
//
// 3 passes, one hex-digit *pair* each (256-pt FWHT) over 64-column LDS tiles
// (~66-68 KB -> 4 blocks/WGP).  Global<->LDS traffic uses CDNA5 async
// direct-to-LDS B128 ops (ASYNCcnt), fully coalesced 256 B rows.  The 192 MB
// L2 keeps the 128 MB working set resident between passes, so HBM sees ~one
// read + one write (~11 us floor at 23.3 TB/s).  1/N folded into the last
// WMMA stage's +/-1 matrix (exact: power of two).
// ---------------------------------------------------------------------------

typedef __attribute__((ext_vector_type(2))) float v2f;
typedef __attribute__((ext_vector_type(8))) float v8f;

#define A_DIM     256                 // transform length per pass
#define COLS      64                  // independent columns per block
#define BLK_ELEMS (A_DIM * COLS)      // 16384 floats per block

// --- CDNA5 async direct-to-LDS ops (ASYNCcnt-tracked) ----------------------
// LDS dest/src operand is a per-lane VGPR holding the LDS *byte* offset.
__device__ __forceinline__ void async_load_b128(unsigned lds_byte, const float* g) {
  asm volatile("global_load_async_to_lds_b128 %0, %1, off"
               :: "v"(lds_byte), "v"((unsigned long long)(uintptr_t)g)
               : "memory");
}
__device__ __forceinline__ void async_store_b128(float* g, unsigned lds_byte) {
  asm volatile("global_store_async_from_lds_b128 %0, %1, off"
               :: "v"((unsigned long long)(uintptr_t)g), "v"(lds_byte)
               : "memory");
}
__device__ __forceinline__ void wait_async0() {
  asm volatile("s_wait_asynccnt 0x0" ::: "memory");
}

// One wave applies H16 (Y = H*X, H[m][k] = +/-s with sign (-1)^popc(m&k))
// to a 16x16 tile in LDS.  SUB=0: a = q*16+k (low digit), SUB=1: a = k*16+q.
// LDS element (a,c) lives at float index a*SA + c*SC.
template <int SUB, int SA, int SC>
__device__ __forceinline__ void h16_tile(float* lds, int tau, int lane,
                                         float sp, float sn) {
  const int n = lane & 15;            // tile column (B/C/D lane index)
  const int h = lane >> 4;            // half-wave select
  const int m = n;                    // A-operand output row (lane&15)
  const int q = tau >> 2;             // fixed hex digit
  const int c = (tau & 3) * 16 + n;   // LDS column

  v8f acc = {0.f, 0.f, 0.f, 0.f, 0.f, 0.f, 0.f, 0.f};
#pragma unroll
  for (int cc = 0; cc < 4; ++cc) {    // K-chunks of 4 (K idx = vgpr + 2*half)
    const int k0 = cc * 4 + 2 * h;
    const int k1 = k0 + 1;
    v2f A, B;
    A.x = (__popc(m & k0) & 1) ? sn : sp;
    A.y = (__popc(m & k1) & 1) ? sn : sp;
    const int a0 = (SUB == 0) ? (q * 16 + k0) : (k0 * 16 + q);
    const int a1 = (SUB == 0) ? (q * 16 + k1) : (k1 * 16 + q);
    B.x = lds[a0 * SA + c * SC];
    B.y = lds[a1 * SA + c * SC];
    acc = __builtin_amdgcn_wmma_f32_16x16x4_f32(false, A, false, B,
                                                (short)0, acc, false, false);
  }
  // D layout: vgpr r -> row m = r + 8*half.  In-place safe: DS ops are
  // in-order per wave, tiles disjoint per wave within a stage.
#pragma unroll
  for (int r = 0; r < 8; ++r) {
    const int mm = r + 8 * h;
    const int aa = (SUB == 0) ? (q * 16 + mm) : (mm * 16 + q);
    lds[aa * SA + c * SC] = acc[r];
  }
}

// MODE 0: a = bits[7:0]   (global stride 1),     c = bits[13:8]  (stride 256)
// MODE 1: a = bits[15:8]  (global stride 256),   c = bits[5:0]   (stride 1)
// MODE 2: a = bits[23:16] (global stride 65536), c = bits[5:0]   (stride 1)
template <int MODE>
__global__ __launch_bounds__(256) void fwht_pass(const float* __restrict__ src,
                                                 float* __restrict__ dst,
                                                 const float scale) {
  // LDS layout: contiguous along the global-contiguous axis so async B128
  // transfers are straight-line; padded strides (x4B multiples of 16).
  constexpr int SA   = (MODE == 0) ? 1 : 68;        // float stride along a
  constexpr int SC   = (MODE == 0) ? 260 : 1;       // float stride along c
  constexpr int NLDS = (MODE == 0) ? COLS * 260 : A_DIM * 68;
  constexpr int AST  = (MODE == 0) ? 1 : (MODE == 1 ? 256 : 65536);
  constexpr int CST  = (MODE == 0) ? 256 : 1;

  __shared__ float lds[NLDS];         // single shared var -> LDS offset 0
  const int t    = threadIdx.x;
  const int lane = t & 31;
  const int wave = t >> 5;
  const int blk  = blockIdx.x;

  int jbase;
  if (MODE == 0)      jbase = blk * BLK_ELEMS;
  else if (MODE == 1) jbase = (blk >> 2) * 65536 + (blk & 3) * 64;
  else                jbase = blk * 64;

  // ------------- global -> LDS: async B128, fully coalesced ---------------
#pragma unroll 4
  for (int i = 0; i < 16; ++i) {
    const int f = i * 256 + t;        // float4 slot
    int a, c;
    if (MODE == 0) { a = (f & 63) * 4; c = f >> 6; }
    else           { c = (f & 15) * 4; a = f >> 4; }
    const unsigned lb = (unsigned)((a * SA + c * SC) * 4);
    async_load_b128(lb, src + jbase + a * AST + c * CST);
  }
  wait_async0();
  __syncthreads();

  // ------- 256-pt FWHT along a = two chained H16 stages via WMMA ----------
  for (int i = 0; i < 8; ++i) h16_tile<0, SA, SC>(lds, wave * 8 + i, lane, 1.0f, -1.0f);
  __syncthreads();
  // fold output scale (1/N on last pass) into the H matrix: exact (pow2)
  for (int i = 0; i < 8; ++i) h16_tile<1, SA, SC>(lds, wave * 8 + i, lane, scale, -scale);
  __syncthreads();

  // ------------- LDS -> global: async B128, fully coalesced ---------------
#pragma unroll 4
  for (int i = 0; i < 16; ++i) {
    const int f = i * 256 + t;
    int a, c;
    if (MODE == 0) { a = (f & 63) * 4; c = f >> 6; }
    else           { c = (f & 15) * 4; a = f >> 4; }
    const unsigned lb = (unsigned)((a * SA + c * SC) * 4);
    async_store_b128(dst + jbase + a * AST + c * CST, lb);
  }
  // S_ENDPGM performs an implicit wait-idle, draining ASYNCcnt.
}

extern "C" void kernel_launch(void* const* d_in, const int* in_sizes, int n_in,
                              void* d_out, int out_size, void* d_ws, size_t ws_size,
                              hipStream_t stream) {
  (void)n_in; (void)d_ws; (void)ws_size; (void)out_size;
  const float* H   = (const float*)d_in[0];
  float*       out = (float*)d_out;
  const int n      = in_sizes[0];           // 1<<24
  const int blocks = n / BLK_ELEMS;         // 1024
  const float inv_n = 1.0f / (float)n;      // exact power of two

  // digits d0,d1 : in -> out (fully contiguous)
  fwht_pass<0><<<blocks, 256, 0, stream>>>(H, out, 1.0f);
  // digits d2,d3 : out -> out in place (disjoint index set per block)
  fwht_pass<1><<<blocks, 256, 0, stream>>>(out, out, 1.0f);
  // digits d4,d5 : out -> out in place, 1/N folded into last WMMA stage
  fwht_pass<2><<<blocks, 256, 0, stream>>>(out, out, inv_n);
}